// GaussianParameters_83064667504940
// MI455X (gfx1250) — compile-verified
//
#include <hip/hip_runtime.h>
#include <math.h>
#include <stdint.h>

#define RED_BLOCKS  1024
#define RED_THREADS 256
#define MAIN_THREADS 256

// 32-bit LDS byte address from a generic shared-memory pointer
// (generic shared address = {aperture_hi32, lds_offset_lo32}; truncation keeps the offset)
__device__ __forceinline__ uint32_t lds_u32(const void* p) {
    return (uint32_t)(uintptr_t)p;
}

// ---------------------------------------------------------------------------
// Kernel 1: per-block partial sums of sum(q*q) over the whole quaternion array
// (reference does qn = q / ||Q||_F, a GLOBAL Frobenius norm). float4 loads,
// deterministic LDS tree reduction.
// ---------------------------------------------------------------------------
__global__ __launch_bounds__(RED_THREADS)
void sumsq_partial_kernel(const float4* __restrict__ q4, int nvec,
                          float* __restrict__ partials) {
    __shared__ float sm[RED_THREADS];
    const int t = threadIdx.x;
    const long long stride = (long long)gridDim.x * RED_THREADS;
    float acc = 0.0f;
    for (long long i = (long long)blockIdx.x * RED_THREADS + t; i < nvec; i += stride) {
        float4 v = q4[i];
        acc += v.x * v.x + v.y * v.y + v.z * v.z + v.w * v.w;
    }
    sm[t] = acc;
    __syncthreads();
    for (int s = RED_THREADS / 2; s > 0; s >>= 1) {
        if (t < s) sm[t] += sm[t + s];
        __syncthreads();
    }
    if (t == 0) partials[blockIdx.x] = sm[0];
}

// ---------------------------------------------------------------------------
// Kernel 2: single block folds the 1024 partials (fixed order) -> ws[0] = 1/norm
// ---------------------------------------------------------------------------
__global__ __launch_bounds__(RED_THREADS)
void finalize_norm_kernel(const float* __restrict__ partials,
                          float* __restrict__ inv_norm_out) {
    __shared__ float sm[RED_THREADS];
    const int t = threadIdx.x;
    float acc = 0.0f;
    for (int i = t; i < RED_BLOCKS; i += RED_THREADS) acc += partials[i];
    sm[t] = acc;
    __syncthreads();
    for (int s = RED_THREADS / 2; s > 0; s >>= 1) {
        if (t < s) sm[t] += sm[t + s];
        __syncthreads();
    }
    if (t == 0) inv_norm_out[0] = 1.0f / sqrtf(sm[0]);
}

// ---------------------------------------------------------------------------
// Kernel 3: main elementwise kernel. One point per thread.
//  - quaternions: coalesced global_load_b128 (float4 per point)
//  - scales: async global->LDS copy (global_load_async_to_lds_b128, ASYNCcnt)
//  - cov / inv_cov: computed in registers, staged in LDS (stride-9 write is
//    bank-conflict-free on 64 banks), streamed out directly from LDS with
//    global_store_async_from_lds_b128 (no VGPR round trip)
// ---------------------------------------------------------------------------
__global__ __launch_bounds__(MAIN_THREADS)
void gaussian_main_kernel(const float4* __restrict__ q4,
                          const float* __restrict__ scales,
                          const float* __restrict__ opac_in,
                          const float* __restrict__ inv_norm_ptr,
                          float* __restrict__ cov_out,
                          float* __restrict__ icov_out,
                          float* __restrict__ opac_out,
                          int n) {
    __shared__ __align__(16) float lds_a[MAIN_THREADS * 9];  // scales staging, then cov
    __shared__ __align__(16) float lds_b[MAIN_THREADS * 9];  // inv_cov

    const int t = threadIdx.x;
    const long long block_base = (long long)blockIdx.x * MAIN_THREADS;
    if (block_base >= n) return;
    const int valid = (block_base + MAIN_THREADS <= (long long)n)
                          ? MAIN_THREADS
                          : (int)((long long)n - block_base);

    // ---- stage this block's scales (3 floats/point) into LDS ----
    {
        const float* sg = scales + block_base * 3;
        const int nsf = valid * 3;
        if ((((uintptr_t)sg) & 15) == 0) {
            const int nv = nsf >> 2;  // full float4s
            const float4* sg4 = (const float4*)sg;
            for (int j = t; j < nv; j += MAIN_THREADS) {
                const uint32_t la = lds_u32(lds_a + 4 * j);
                const float4* ga = sg4 + j;
                // async DMA: global -> LDS, 16 bytes per lane, tracked by ASYNCcnt
                asm volatile("global_load_async_to_lds_b128 %0, %1, off"
                             :: "v"(la), "v"(ga)
                             : "memory");
            }
            for (int j = (nv << 2) + t; j < nsf; j += MAIN_THREADS) lds_a[j] = sg[j];
            asm volatile("s_wait_asynccnt 0x0" ::: "memory");
        } else {
            for (int j = t; j < nsf; j += MAIN_THREADS) lds_a[j] = sg[j];
        }
    }
    __syncthreads();

    float sx = 0.0f, sy = 0.0f, sz = 0.0f;
    float4 qv = make_float4(0.0f, 0.0f, 0.0f, 0.0f);
    if (t < valid) {
        sx = lds_a[t * 3 + 0];
        sy = lds_a[t * 3 + 1];
        sz = lds_a[t * 3 + 2];
        qv = q4[block_base + t];
    }
    __syncthreads();  // lds_a now free to be reused for cov

    const float inv_norm = inv_norm_ptr[0];  // uniform -> scalar load

    if (t < valid) {
        const float w = qv.x * inv_norm;
        const float x = qv.y * inv_norm;
        const float y = qv.z * inv_norm;
        const float z = qv.w * inv_norm;

        const float xx = x * x, yy = y * y, zz = z * z;
        const float xy = x * y, xz = x * z, yz = y * z;
        const float xw = x * w, yw = y * w, zw = z * w;

        const float r00 = 1.0f - 2.0f * (yy + zz);
        const float r01 = 2.0f * (xy - zw);
        const float r02 = 2.0f * (xz + yw);
        const float r10 = 2.0f * (xy + zw);
        const float r11 = 1.0f - 2.0f * (xx + zz);
        const float r12 = 2.0f * (yz - xw);
        const float r20 = 2.0f * (xz - yw);
        const float r21 = 2.0f * (yz + xw);
        const float r22 = 1.0f - 2.0f * (xx + yy);

        const float e0 = expf(2.0f * sx), e1 = expf(2.0f * sy), e2 = expf(2.0f * sz);
        const float f0 = expf(-2.0f * sx), f1 = expf(-2.0f * sy), f2 = expf(-2.0f * sz);

        // cov = R diag(e) R^T  (symmetric, 6 unique entries)
        {
            const float a0 = r00 * e0, a1 = r01 * e1, a2 = r02 * e2;
            const float b0 = r10 * e0, b1 = r11 * e1, b2 = r12 * e2;
            const float c0 = r20 * e0, c1 = r21 * e1, c2 = r22 * e2;
            const float m00 = a0 * r00 + a1 * r01 + a2 * r02;
            const float m01 = a0 * r10 + a1 * r11 + a2 * r12;
            const float m02 = a0 * r20 + a1 * r21 + a2 * r22;
            const float m11 = b0 * r10 + b1 * r11 + b2 * r12;
            const float m12 = b0 * r20 + b1 * r21 + b2 * r22;
            const float m22 = c0 * r20 + c1 * r21 + c2 * r22;
            float* L = lds_a + t * 9;  // stride 9 (odd) -> conflict-free mod 64
            L[0] = m00; L[1] = m01; L[2] = m02;
            L[3] = m01; L[4] = m11; L[5] = m12;
            L[6] = m02; L[7] = m12; L[8] = m22;
        }
        // inv_cov = R diag(f) R^T
        {
            const float a0 = r00 * f0, a1 = r01 * f1, a2 = r02 * f2;
            const float b0 = r10 * f0, b1 = r11 * f1, b2 = r12 * f2;
            const float c0 = r20 * f0, c1 = r21 * f1, c2 = r22 * f2;
            const float m00 = a0 * r00 + a1 * r01 + a2 * r02;
            const float m01 = a0 * r10 + a1 * r11 + a2 * r12;
            const float m02 = a0 * r20 + a1 * r21 + a2 * r22;
            const float m11 = b0 * r10 + b1 * r11 + b2 * r12;
            const float m12 = b0 * r20 + b1 * r21 + b2 * r22;
            const float m22 = c0 * r20 + c1 * r21 + c2 * r22;
            float* L = lds_b + t * 9;
            L[0] = m00; L[1] = m01; L[2] = m02;
            L[3] = m01; L[4] = m11; L[5] = m12;
            L[6] = m02; L[7] = m12; L[8] = m22;
        }

        // opacity: coalesced b32 load + store
        const float o = opac_in[block_base + t];
        opac_out[block_base + t] = 1.0f / (1.0f + expf(-o));
    }
    __syncthreads();  // s_wait_dscnt + barrier: LDS results visible block-wide

    // ---- stream cov / inv_cov out of LDS (async LDS->global, 16B per lane) ----
    const long long obase = block_base * 9;  // blockIdx*2304 floats -> 16B-aligned offset
    const int nfl = valid * 9;
    float* cg = cov_out + obase;
    float* ig = icov_out + obase;
    if (((((uintptr_t)cg) | ((uintptr_t)ig)) & 15) == 0) {
        const int nv = nfl >> 2;  // full float4s
        for (int j = t; j < nv; j += MAIN_THREADS) {
            const uint32_t lca = lds_u32(lds_a + 4 * j);
            const uint32_t lia = lds_u32(lds_b + 4 * j);
            const float4* gc = (const float4*)cg + j;
            const float4* gi = (const float4*)ig + j;
            asm volatile("global_store_async_from_lds_b128 %0, %1, off"
                         :: "v"(gc), "v"(lca)
                         : "memory");
            asm volatile("global_store_async_from_lds_b128 %0, %1, off"
                         :: "v"(gi), "v"(lia)
                         : "memory");
        }
        for (int j = (nv << 2) + t; j < nfl; j += MAIN_THREADS) {
            cg[j] = lds_a[j];
            ig[j] = lds_b[j];
        }
        // outstanding async stores must complete before LDS is reused/freed;
        // S_ENDPGM also does an implicit wait-idle, this is belt-and-braces.
        asm volatile("s_wait_asynccnt 0x0" ::: "memory");
    } else {
        for (int j = t; j < nfl; j += MAIN_THREADS) {
            cg[j] = lds_a[j];
            ig[j] = lds_b[j];
        }
    }
}

// ---------------------------------------------------------------------------
extern "C" void kernel_launch(void* const* d_in, const int* in_sizes, int n_in,
                              void* d_out, int out_size, void* d_ws, size_t ws_size,
                              hipStream_t stream) {
    const float* q  = (const float*)d_in[0];   // (N,4)
    const float* sc = (const float*)d_in[1];   // (N,3)
    const float* op = (const float*)d_in[2];   // (N,1)
    const int n = in_sizes[0] / 4;

    float* out  = (float*)d_out;
    float* cov  = out;                          // N*9
    float* icov = out + (long long)9 * n;       // N*9
    float* opa  = out + (long long)18 * n;      // N

    float* wsf      = (float*)d_ws;
    float* inv_norm = wsf;        // ws[0]
    float* partials = wsf + 16;   // ws[16 .. 16+RED_BLOCKS)

    const int nvec4 = in_sizes[0] / 4;  // float4 count of quaternion array (== n)

    sumsq_partial_kernel<<<RED_BLOCKS, RED_THREADS, 0, stream>>>(
        (const float4*)q, nvec4, partials);
    finalize_norm_kernel<<<1, RED_THREADS, 0, stream>>>(partials, inv_norm);

    const int blocks = (n + MAIN_THREADS - 1) / MAIN_THREADS;
    gaussian_main_kernel<<<blocks, MAIN_THREADS, 0, stream>>>(
        (const float4*)q, sc, op, inv_norm, cov, icov, opa, n);
}